// MDPPInitEmbedding_37752762532202
// MI455X (gfx1250) — compile-verified
//
#include <hip/hip_runtime.h>
#include <hip/hip_bf16.h>
#include <math.h>

#define B_ 8
#define N_ 4096
#define E_ 256
#define ASTRIDE 516   // 512 + 4 pad: lanes m=0..15 hit distinct LDS banks

typedef __attribute__((ext_vector_type(2))) float v2f;
typedef __attribute__((ext_vector_type(8))) float v8f;

// ---------------------------------------------------------------------------
// Kernel 1: compact probe points per batch: probe_locs[b][0..cnt) = (x,y)
// ---------------------------------------------------------------------------
__global__ __launch_bounds__(256) void probe_compact_kernel(
    const float* __restrict__ locs, const unsigned char* __restrict__ probe,
    float* __restrict__ px, float* __restrict__ py, int* __restrict__ pcnt) {
  int b = blockIdx.x;
  __shared__ int counter;
  if (threadIdx.x == 0) counter = 0;
  __syncthreads();
  for (int n = threadIdx.x; n < N_; n += blockDim.x) {
    if (probe[b * N_ + n] != 0) {
      int idx = atomicAdd(&counter, 1);
      px[b * N_ + idx] = locs[(b * N_ + n) * 2 + 0];
      py[b * N_ + idx] = locs[(b * N_ + n) * 2 + 1];
    }
  }
  __syncthreads();
  if (threadIdx.x == 0) pcnt[b] = counter;
}

// ---------------------------------------------------------------------------
// Kernel 2: coefficient GEMM with fp32 WMMA.
//   A (16x512, rows 0..3 used, 4..15 zero):
//     row0 = [W1[0,:], 0]   row1 = [W1[1,:], 0]
//     row2 = [0, W2[0,:]]   row3 = [b1, b2]
//   U (4x256) = A @ W3 ; U[3] += b3
// A is staged into LDS once (branch-free), so the K loop is a straight
// ds_load_b64 (A frag) + 2x global_load_b32 (B frag) + v_wmma chain.
// A-matrix 16x4 f32 layout: lane%16 = M; VGPR j + 2*(lane/16) = K offset.
// B-matrix 4x16 f32 layout: lane%16 = N; VGPR j + 2*(lane/16) = K offset.
// C/D 16x16 f32 layout: VGPR r -> row r (lanes 0-15) / r+8 (lanes 16-31).
// ---------------------------------------------------------------------------
__global__ __launch_bounds__(512) void coef_wmma_kernel(
    const float* __restrict__ W1, const float* __restrict__ b1,
    const float* __restrict__ W2, const float* __restrict__ b2,
    const float* __restrict__ W3, const float* __restrict__ b3,
    float* __restrict__ U) {
  __shared__ float As[16 * ASTRIDE];   // ~33 KB of the 320 KB WGP LDS

  // Cooperative, branch-free fill: thread t owns column k = t of all 16 rows.
  {
    int k = threadIdx.x;               // 0..511
    float lo = (k < E_) ? 1.0f : 0.0f; // k in first half (node_emb part)
    float r0 = (k < E_) ? W1[k & (E_ - 1)] : 0.0f;
    float r1 = (k < E_) ? W1[E_ + (k & (E_ - 1))] : 0.0f;
    float r2 = (k >= E_) ? W2[k & (E_ - 1)] : 0.0f;
    float r3 = (k < E_) ? b1[k & (E_ - 1)] : b2[k & (E_ - 1)];
    (void)lo;
    As[0 * ASTRIDE + k] = r0;
    As[1 * ASTRIDE + k] = r1;
    As[2 * ASTRIDE + k] = r2;
    As[3 * ASTRIDE + k] = r3;
#pragma unroll
    for (int r = 4; r < 16; ++r) As[r * ASTRIDE + k] = 0.0f;
  }
  __syncthreads();

  int wave = threadIdx.x >> 5;         // 16 waves, one 16-col tile of N each
  int lane = threadIdx.x & 31;
  int n0   = wave * 16;
  int m    = lane & 15;
  int khalf = (lane >> 4) << 1;        // 0 for lanes 0-15, 2 for lanes 16-31

  const float* arow = As + m * ASTRIDE;
  const float* bcol = W3 + n0 + m;

  v8f c = {};
#pragma unroll 4
  for (int kb = 0; kb < 2 * E_; kb += 4) {
    int k0 = kb + khalf;
    v2f a, bb;
    a.x  = arow[k0 + 0];               // one ds_load_b64 (consecutive floats)
    a.y  = arow[k0 + 1];
    bb.x = bcol[(k0 + 0) * E_];
    bb.y = bcol[(k0 + 1) * E_];
    c = __builtin_amdgcn_wmma_f32_16x16x4_f32(
        /*neg_a=*/false, a, /*neg_b=*/false, bb,
        /*c_mod=*/(short)0, c, /*reuse_a=*/false, /*reuse_b=*/false);
  }
  if (lane < 16) {
#pragma unroll
    for (int r = 0; r < 4; ++r) {      // rows 0..3 live in VGPRs 0..3, lanes 0-15
      float v = c[r];
      if (r == 3) v += b3[n0 + lane];
      U[r * E_ + n0 + lane] = v;
    }
  }
}

// ---------------------------------------------------------------------------
// Kernel 3: min distance to any probe point, tiled through LDS.
// ---------------------------------------------------------------------------
__global__ __launch_bounds__(256) void min_dist_kernel(
    const float* __restrict__ locs, const float* __restrict__ px,
    const float* __restrict__ py, const int* __restrict__ pcnt,
    float* __restrict__ md) {
  int b = blockIdx.y;
  int n = blockIdx.x * 256 + threadIdx.x;
  float x = locs[(b * N_ + n) * 2 + 0];
  float y = locs[(b * N_ + n) * 2 + 1];
  int cnt = pcnt[b];                 // >= 1 (probe[:,0] forced true)
  __shared__ float sx[256], sy[256];
  float best = 3.402823466e38f;
  for (int base = 0; base < cnt; base += 256) {
    int i = base + (int)threadIdx.x;
    if (i < cnt) {
      sx[threadIdx.x] = px[b * N_ + i];
      sy[threadIdx.x] = py[b * N_ + i];
    }
    __syncthreads();
    int mlim = (cnt - base) < 256 ? (cnt - base) : 256;
    for (int j = 0; j < mlim; ++j) {
      float dx = x - sx[j];
      float dy = y - sy[j];
      float d2 = fmaf(dx, dx, dy * dy);
      best = fminf(best, d2);
    }
    __syncthreads();
  }
  md[b * N_ + n] = sqrtf(best);      // min(sqrt(d2)) == sqrt(min(d2))
}

// ---------------------------------------------------------------------------
// Kernel 4: streaming epilogue: out[b,n,e] = x*U0[e] + y*U1[e] + md*U2[e] + U3[e]
// float4 stores; this is the 33.5 MB HBM-bound stage.
// ---------------------------------------------------------------------------
__global__ __launch_bounds__(256) void out_stream_kernel(
    const float* __restrict__ locs, const float* __restrict__ md,
    const float* __restrict__ U, float* __restrict__ out) {
  __shared__ float sx[64], sy[64], sm[64];
  int b  = blockIdx.x >> 6;          // 64 blocks per batch
  int n0 = (blockIdx.x & 63) * 64;   // 64 points per block
  int t  = threadIdx.x;
  if (t < 64) {
    int n = n0 + t;
    sx[t] = locs[(b * N_ + n) * 2 + 0];
    sy[t] = locs[(b * N_ + n) * 2 + 1];
    sm[t] = md[b * N_ + n];
  }
  __syncthreads();
  int eg = (t & 63) * 4;             // 4 consecutive e per thread
  const float4 u0 = *reinterpret_cast<const float4*>(U + 0 * E_ + eg);
  const float4 u1 = *reinterpret_cast<const float4*>(U + 1 * E_ + eg);
  const float4 u2 = *reinterpret_cast<const float4*>(U + 2 * E_ + eg);
  const float4 u3 = *reinterpret_cast<const float4*>(U + 3 * E_ + eg);
  for (int nn = t >> 6; nn < 64; nn += 4) {
    float x = sx[nn], y = sy[nn], d = sm[nn];
    float4 o;
    o.x = fmaf(x, u0.x, fmaf(y, u1.x, fmaf(d, u2.x, u3.x)));
    o.y = fmaf(x, u0.y, fmaf(y, u1.y, fmaf(d, u2.y, u3.y)));
    o.z = fmaf(x, u0.z, fmaf(y, u1.z, fmaf(d, u2.z, u3.z)));
    o.w = fmaf(x, u0.w, fmaf(y, u1.w, fmaf(d, u2.w, u3.w)));
    *reinterpret_cast<float4*>(out + (size_t)(b * N_ + n0 + nn) * E_ + eg) = o;
  }
}

// ---------------------------------------------------------------------------
// Launch
// ---------------------------------------------------------------------------
extern "C" void kernel_launch(void* const* d_in, const int* in_sizes, int n_in,
                              void* d_out, int out_size, void* d_ws, size_t ws_size,
                              hipStream_t stream) {
  (void)in_sizes; (void)n_in; (void)out_size; (void)ws_size;
  const float*         locs  = (const float*)d_in[0];
  const unsigned char* probe = (const unsigned char*)d_in[1];  // jax bool, 1B/elem
  const float*         W1    = (const float*)d_in[2];
  const float*         b1    = (const float*)d_in[3];
  const float*         W2    = (const float*)d_in[4];
  const float*         b2    = (const float*)d_in[5];
  const float*         W3    = (const float*)d_in[6];
  const float*         b3    = (const float*)d_in[7];
  float* out = (float*)d_out;

  // Workspace layout (floats): px[B*N], py[B*N], md[B*N], U[4*E], pcnt[B] ints
  float* wsf = (float*)d_ws;
  float* px  = wsf;
  float* py  = wsf + (size_t)B_ * N_;
  float* md  = wsf + (size_t)2 * B_ * N_;
  float* U   = wsf + (size_t)3 * B_ * N_;
  int*   pcnt = (int*)(wsf + (size_t)3 * B_ * N_ + 4 * E_);

  probe_compact_kernel<<<B_, 256, 0, stream>>>(locs, probe, px, py, pcnt);
  coef_wmma_kernel<<<1, 512, 0, stream>>>(W1, b1, W2, b2, W3, b3, U);
  min_dist_kernel<<<dim3(N_ / 256, B_), 256, 0, stream>>>(locs, px, py, pcnt, md);
  out_stream_kernel<<<B_ * (N_ / 64), 256, 0, stream>>>(locs, md, U, out);
}